// Memory_17832704213565
// MI455X (gfx1250) — compile-verified
//
#include <hip/hip_runtime.h>
#include <math.h>

typedef float v2f __attribute__((ext_vector_type(2)));
typedef float v8f __attribute__((ext_vector_type(8)));

#define BB 64
#define NN 16384
#define MM 64
#define BN (BB * NN)          // 1048576
#define COS_EPS 1e-8f
#define NORM_EPS 1e-16f

// ---------------- kernel 0: zero the atomic accumulators ----------------
__global__ void k_zero(float* __restrict__ sums, int n) {
    int i = blockIdx.x * blockDim.x + threadIdx.x;
    if (i < n) sums[i] = 0.0f;
}

// ---------------- kernel 1: content addressing ----------------
// One wave per (batch, 16-row tile).
// Stage the 4KB tile global->LDS with 8 fully-coalesced
// global_load_async_to_lds_b128 (512B contiguous per issue, ASYNCcnt),
// then feed WMMA from LDS in the lane->row layout:
//   A = key chunk broadcast over rows (16x4), B = memory^T chunk (4x16)
//   -> D[i][j] = dot[n0+j] constant over i, so every lane's c[0] is the
//      dot product of its own row. Row/key sum-of-squares ride along on
//      the same float2 values; lane L and L^16 cover complementary K
//      halves, merged with one shfl_xor.
__global__ __launch_bounds__(256) void k_content(
    const float* __restrict__ key, const float* __restrict__ beta,
    const float* __restrict__ mem, float* __restrict__ e,
    float* __restrict__ sumE)
{
    constexpr int RS = 72;                    // padded LDS row stride (floats):
                                              // 288B = 16B-aligned writes,
                                              // 8B-aligned b64 reads, <=2-way banks
    __shared__ float tileBuf[8 * 16 * RS];    // 8 waves x padded 16x64 tile (36KB)

    const int lane = threadIdx.x & 31;
    const int wv   = threadIdx.x >> 5;
    const int wid  = blockIdx.x * 8 + wv;
    const int b    = wid >> 10;               // NN/16 = 1024 tiles per batch
    const int n0   = (wid & 1023) << 4;
    const int row  = lane & 15;
    const int off  = (lane >> 4) << 1;        // 0 for lanes 0-15, 2 for 16-31

    float* s = &tileBuf[wv * 16 * RS];
    const float* __restrict__ tileG = mem + ((size_t)b * NN + n0) * MM;
    const unsigned sBase = (unsigned)(uintptr_t)s;   // LDS offset = low 32 bits

    // ---- async stage: 8 x 512B contiguous, lanes map ALONG the rows ----
#pragma unroll
    for (int j = 0; j < 8; ++j) {
        const int r = 2 * j + (lane >> 4);               // dest row in tile
        unsigned ldsb = sBase + r * (RS * 4) + (lane & 15) * 16;
        unsigned long long g =
            (unsigned long long)(uintptr_t)(tileG + j * 128 + lane * 4);
        asm volatile("global_load_async_to_lds_b128 %0, %1, off"
                     :: "v"(ldsb), "v"(g) : "memory");
    }

    const float* __restrict__ keyB = key + b * MM;

    asm volatile("s_wait_asynccnt 0x0" ::: "memory");

    // ---- WMMA accumulation over K = 64 (16 chunks of 4) ----
    const float* srow = s + row * RS;
    v8f c = {};
    float mss = 0.0f, kss = 0.0f;
#pragma unroll
    for (int kb = 0; kb < MM; kb += 4) {
        v2f bv = *(const v2f*)(srow + kb + off);   // ds_load_b64, WMMA layout
        v2f av;
        av.x = keyB[kb + off];
        av.y = keyB[kb + off + 1];
        mss += bv.x * bv.x + bv.y * bv.y;
        kss += av.x * av.x + av.y * av.y;
        c = __builtin_amdgcn_wmma_f32_16x16x4_f32(
                false, av, false, bv, (short)0, c, false, false);
    }
    // merge complementary K halves (lane ^ 16 handles same row)
    mss += __shfl_xor(mss, 16, 32);
    kss += __shfl_xor(kss, 16, 32);

    const float dot   = c[0];                      // dot[n0 + row], all lanes
    const float denom = fmaxf(sqrtf(kss) * sqrtf(mss), COS_EPS);
    const float ev    = __expf(beta[b] * (dot / denom));

    // wave-reduce partial softmax denominator (rows counted once: lanes 0-15)
    float part = (lane < 16) ? ev : 0.0f;
#pragma unroll
    for (int o = 16; o >= 1; o >>= 1) part += __shfl_xor(part, o, 32);
    if (lane == 0) atomicAdd(&sumE[b], part);
    if (lane < 16) e[(size_t)b * NN + n0 + lane] = ev;
}

// ---------------- kernel 2: softmax finish + interpolation gate ---------
__global__ __launch_bounds__(256) void k_interp(
    const float* __restrict__ e, const float* __restrict__ sumE,
    const float* __restrict__ gate, const float* __restrict__ la,
    float* __restrict__ wg)
{
    int i = blockIdx.x * blockDim.x + threadIdx.x;   // [0, B*N)
    int b = i >> 14;
    float g  = gate[b];
    float wc = e[i] / sumE[b];
    wg[i] = g * wc + (1.0f - g) * la[i];
}

// ---------------- kernel 3: circular 3-tap shift + sharpen + partial sum
__global__ __launch_bounds__(256) void k_shift_pow(
    const float* __restrict__ wg, const float* __restrict__ shift,
    const float* __restrict__ sharpen, float* __restrict__ wp,
    float* __restrict__ sumP)
{
    int i = blockIdx.x * blockDim.x + threadIdx.x;
    int b = i >> 14;
    int n = i & (NN - 1);
    size_t base = (size_t)b << 14;
    int nm = (n - 1) & (NN - 1);
    int np = (n + 1) & (NN - 1);
    // ws[n] = s0*w[n-1] + s1*w[n] + s2*w[n+1]  (circular cross-correlation)
    float ws = shift[b * 3 + 0] * wg[base + nm]
             + shift[b * 3 + 1] * wg[base + n]
             + shift[b * 3 + 2] * wg[base + np];
    float p = __powf(ws, sharpen[b]);
    wp[i] = p;
    // per-wave partial sum (N divisible by 256 -> wave never straddles b)
#pragma unroll
    for (int o = 16; o >= 1; o >>= 1) p += __shfl_xor(p, o, 32);
    if ((threadIdx.x & 31) == 0) atomicAdd(&sumP[b], p);
}

// ---------------- kernel 4: renormalize --------------------------------
__global__ __launch_bounds__(256) void k_norm(
    const float* __restrict__ wp, const float* __restrict__ sumP,
    float* __restrict__ out)
{
    int i = blockIdx.x * blockDim.x + threadIdx.x;
    int b = i >> 14;
    out[i] = wp[i] / (sumP[b] + NORM_EPS);
}

extern "C" void kernel_launch(void* const* d_in, const int* in_sizes, int n_in,
                              void* d_out, int out_size, void* d_ws, size_t ws_size,
                              hipStream_t stream) {
    const float* key     = (const float*)d_in[0];   // [B, M]
    const float* beta    = (const float*)d_in[1];   // [B, 1]
    const float* gate    = (const float*)d_in[2];   // [B, 1]
    const float* shift   = (const float*)d_in[3];   // [B, 3]
    const float* sharpen = (const float*)d_in[4];   // [B, 1]
    const float* la      = (const float*)d_in[5];   // [B, N]
    const float* mem     = (const float*)d_in[6];   // [B, N, M]
    float* out = (float*)d_out;

    float* ws   = (float*)d_ws;
    float* e    = ws;              // [B*N]  exp(beta*cos); reused as wp
    float* wg   = ws + BN;         // [B*N]  gated weights
    float* sumE = ws + 2 * BN;     // [B]
    float* sumP = sumE + BB;       // [B]
    float* wp   = e;               // reuse: e dead after k_interp

    k_zero<<<1, 128, 0, stream>>>(sumE, 2 * BB);

    // B * (N/16) waves, 8 waves per 256-thread block
    k_content<<<(BB * (NN / 16)) / 8, 256, 0, stream>>>(key, beta, mem, e, sumE);

    const int blocks = BN / 256;
    k_interp   <<<blocks, 256, 0, stream>>>(e, sumE, gate, la, wg);
    k_shift_pow<<<blocks, 256, 0, stream>>>(wg, shift, sharpen, wp, sumP);
    k_norm     <<<blocks, 256, 0, stream>>>(wp, sumP, out);
}